// Attn_61881888800871
// MI455X (gfx1250) — compile-verified
//
#include <hip/hip_runtime.h>
#include <hip/hip_bf16.h>
#include <math.h>

// Problem dims (match reference)
#define BATCH 32
#define SEQ   4096
#define HID   1024
#define D2    2048   // 2*H, K-depth of the streaming dot and N of the v-GEMM

typedef float v2f  __attribute__((ext_vector_type(2)));
typedef float v4f  __attribute__((ext_vector_type(4)));
typedef float v8f  __attribute__((ext_vector_type(8)));

// ---------------------------------------------------------------------------
// Kernel 1: v[b,d] = sum_h hidden[b,h] * W[h,d]
//   GEMM  M=BATCH(32) x N=D2(2048) x K=HID(1024), fp32, exact precision.
//   One wave (32 threads) per 16x16 C tile, V_WMMA_F32_16X16X4_F32 over K.
//
//   Fragment layouts per CDNA5 ISA 7.12.2:
//     A 16x4 f32 : lanes 0-15 -> (M=lane, K=k+0 / k+1) in vgpr0/1
//                  lanes 16-31 -> (M=lane-16, K=k+2 / k+3)
//     B 4x16 f32 : lanes 0-15 -> (K=k+0 / k+1, N=n0+lane) in vgpr0/1
//                  lanes 16-31 -> (K=k+2 / k+3, N=n0+lane-16)
//     C/D 16x16  : vgpr r, lanes 0-15 -> (M=r,   N=lane)
//                          lanes16-31 -> (M=r+8, N=lane-16)
// ---------------------------------------------------------------------------
__global__ __launch_bounds__(32) void v_gemm_wmma(const float* __restrict__ hidden,
                                                  const float* __restrict__ W,
                                                  float* __restrict__ v) {
  const int lane = threadIdx.x;        // 0..31, EXEC all ones (WMMA requirement)
  const int l    = lane & 15;
  const int half = lane >> 4;          // 0 or 1
  const int n0   = blockIdx.x * 16;    // N tile (0..2047)
  const int m0   = blockIdx.y * 16;    // M tile (0 or 16)

  v8f acc = {};

  // Per-lane base pointers; kk offset 2*half selects the K pair this lane owns.
  const float* arow = hidden + (size_t)(m0 + l) * HID + 2 * half;
  const float* bcol = W + (size_t)(2 * half) * D2 + (n0 + l);

  for (int k = 0; k < HID; k += 4) {
    v2f a, b;
    a.x = arow[k];
    a.y = arow[k + 1];
    b.x = bcol[(size_t)k * D2];
    b.y = bcol[(size_t)(k + 1) * D2];
    acc = __builtin_amdgcn_wmma_f32_16x16x4_f32(
        /*neg_a=*/false, a, /*neg_b=*/false, b,
        /*c_mod=*/(short)0, acc, /*reuse_a=*/false, /*reuse_b=*/false);
  }

  float* crow = v + (size_t)(m0 + 8 * half) * D2 + (n0 + l);
#pragma unroll
  for (int r = 0; r < 8; ++r) crow[(size_t)r * D2] = acc[r];
}

// ---------------------------------------------------------------------------
// Kernel 2: energies[b,s] = v[b,:] . enc[b,s,:]   (the 1-GiB streaming phase)
//   256 threads = 8 waves per block; one wave per sequence row.
//   enc is streamed once -> non-temporal b128 loads (keep 1 GiB out of L2).
//   v[b] is 8 KB reused 4096x -> regular loads, lives in WGP$/L2.
// ---------------------------------------------------------------------------
__global__ __launch_bounds__(256) void energies_stream(const float* __restrict__ enc,
                                                       const float* __restrict__ v,
                                                       float* __restrict__ e) {
  const int wave = threadIdx.x >> 5;           // 0..7
  const int lane = threadIdx.x & 31;
  const int b    = blockIdx.y;
  const int s    = blockIdx.x * 8 + wave;

  const v4f* erow = (const v4f*)(enc + ((size_t)b * SEQ + s) * D2);
  const v4f* vrow = (const v4f*)(v + (size_t)b * D2);

  float acc = 0.0f;
#pragma unroll
  for (int i = 0; i < D2 / (32 * 4); ++i) {     // 16 iterations of b128 per lane
    v4f x = __builtin_nontemporal_load(&erow[i * 32 + lane]);
    v4f y = vrow[i * 32 + lane];
    acc = fmaf(x.x, y.x, acc);
    acc = fmaf(x.y, y.y, acc);
    acc = fmaf(x.z, y.z, acc);
    acc = fmaf(x.w, y.w, acc);
  }

  // wave32 butterfly reduction
#pragma unroll
  for (int off = 16; off > 0; off >>= 1) acc += __shfl_xor(acc, off, 32);

  if (lane == 0) e[(size_t)b * SEQ + s] = acc;
}

// ---------------------------------------------------------------------------
// Kernel 3: out[b,0,:] = softmax(energies[b,:]) over S=4096.
//   One block (256 threads, 8 waves) per batch row. The hidden·bias constant
//   was dropped: softmax is shift-invariant, so it has no effect.
// ---------------------------------------------------------------------------
__global__ __launch_bounds__(256) void softmax_rows(const float* __restrict__ e,
                                                    float* __restrict__ out) {
  __shared__ float red[8];
  const int b   = blockIdx.x;
  const int tid = threadIdx.x;
  const int lane = tid & 31;
  const int wave = tid >> 5;
  const float* row = e + (size_t)b * SEQ;

  float vals[SEQ / 256];
  float m = -INFINITY;
#pragma unroll
  for (int i = 0; i < SEQ / 256; ++i) {
    vals[i] = row[tid + i * 256];
    m = fmaxf(m, vals[i]);
  }
#pragma unroll
  for (int off = 16; off > 0; off >>= 1) m = fmaxf(m, __shfl_xor(m, off, 32));
  if (lane == 0) red[wave] = m;
  __syncthreads();
  float bm = red[0];
#pragma unroll
  for (int w = 1; w < 8; ++w) bm = fmaxf(bm, red[w]);
  __syncthreads();

  float sum = 0.0f;
#pragma unroll
  for (int i = 0; i < SEQ / 256; ++i) {
    vals[i] = __expf(vals[i] - bm);
    sum += vals[i];
  }
#pragma unroll
  for (int off = 16; off > 0; off >>= 1) sum += __shfl_xor(sum, off, 32);
  if (lane == 0) red[wave] = sum;
  __syncthreads();
  float bs = 0.0f;
#pragma unroll
  for (int w = 0; w < 8; ++w) bs += red[w];
  const float inv = 1.0f / bs;

  float* orow = out + (size_t)b * SEQ;
#pragma unroll
  for (int i = 0; i < SEQ / 256; ++i) orow[tid + i * 256] = vals[i] * inv;
}

// ---------------------------------------------------------------------------
// Launch: inputs are (hidden[B,H], encoder_outputs[B,S,2H], W[H,2H], bias[H])
// all fp32. bias is intentionally unused (cancels in softmax). Output [B,1,S].
// Workspace: v (32*2048 f32 = 256 KB) then energies (32*4096 f32 = 512 KB).
// ---------------------------------------------------------------------------
extern "C" void kernel_launch(void* const* d_in, const int* in_sizes, int n_in,
                              void* d_out, int out_size, void* d_ws, size_t ws_size,
                              hipStream_t stream) {
  const float* hidden = (const float*)d_in[0];
  const float* enc    = (const float*)d_in[1];
  const float* W      = (const float*)d_in[2];
  // d_in[3] = bias, unused: softmax(e + const) == softmax(e)

  float* out = (float*)d_out;
  float* v   = (float*)d_ws;                    // [BATCH, D2]
  float* e   = v + (size_t)BATCH * D2;          // [BATCH, SEQ]

  // Phase 1: v = hidden @ W  (fp32 WMMA, 128 N-tiles x 2 M-tiles, 1 wave each)
  v_gemm_wmma<<<dim3(D2 / 16, BATCH / 16), 32, 0, stream>>>(hidden, W, v);

  // Phase 2: 1 GiB stream, bandwidth-bound (~46 us at 23.3 TB/s)
  energies_stream<<<dim3(SEQ / 8, BATCH), 256, 0, stream>>>(enc, v, e);

  // Phase 3: softmax per batch row
  softmax_rows<<<BATCH, 256, 0, stream>>>(e, out);
}